// BidirectionalConvMambaBlock_67740224192687
// MI455X (gfx1250) — compile-verified
//
#include <hip/hip_runtime.h>
#include <math.h>

#define DIM     256
#define DSTATE  32
#define DINNER  512
#define DTRANK  16
#define BATCH   8
#define SEQ     2048
#define NT      (BATCH*SEQ)   /* 16384 tokens */

typedef float v2f __attribute__((ext_vector_type(2)));
typedef float v8f __attribute__((ext_vector_type(8)));

__device__ __forceinline__ float silu_f(float x) { return x / (1.f + __expf(-x)); }

enum { EPI_NONE = 0, EPI_BIAS = 1, EPI_BIAS_GELU = 2, EPI_BIAS_SOFTPLUS = 3 };

// ---------------------------------------------------------------------------
// Register-blocked fp32 GEMM on the CDNA5 matrix pipe:
//   C[M,N] = A[M,K] @ W[N,K]^T
// One wave computes a (16*BM) x (16*BN) block: BM x BN accumulators of
// V_WMMA_F32_16X16X4_F32 tiles; each A/B fragment is reused BN/BM times,
// giving BM*BN wmmas per (BM+BN) float2 loads per K-step.
// A operand layout (ISA 7.12.2): lane l -> row M = l%16, K = 2*(l/16)+{0,1}
// B operand layout (mirrored):   lane l -> col N = l%16, K = 2*(l/16)+{0,1}
// C/D layout: VGPR q -> row m0 + q + 8*(l/16), col n0 + l%16
// ---------------------------------------------------------------------------
template<int EPI, int BM, int BN>
__global__ __launch_bounds__(256)
void gemm_wmma_f32(const float* __restrict__ A, int lda,
                   const float* __restrict__ W, int ldw,
                   const float* __restrict__ bias,
                   float* __restrict__ C, int ldc,
                   int M, int N, int K)
{
    const int lane = threadIdx.x & 31;
    const int wave = blockIdx.x * (blockDim.x >> 5) + (threadIdx.x >> 5);
    const int ntn  = N / (16 * BN);
    const int ntm  = M / (16 * BM);
    if (wave >= ntm * ntn) return;          // wave-uniform guard: EXEC stays all-1s
    const int m0 = (wave / ntn) * (16 * BM);
    const int n0 = (wave % ntn) * (16 * BN);
    const int r  = lane & 15;
    const int kh = (lane >> 4) << 1;        // 0 or 2

    const float* ap = A + (size_t)(m0 + r) * lda + kh;
    const float* wp = W + (size_t)(n0 + r) * ldw + kh;

    v8f acc[BM][BN] = {};

    for (int k = 0; k < K; k += 4) {
        v2f a[BM], b[BN];
        #pragma unroll
        for (int i = 0; i < BM; ++i)
            a[i] = *(const v2f*)(ap + (size_t)i * 16 * lda + k);
        #pragma unroll
        for (int j = 0; j < BN; ++j)
            b[j] = *(const v2f*)(wp + (size_t)j * 16 * ldw + k);
        if (k + 64 < K) {                   // gfx1250 global_prefetch_b8 on streaming A
            __builtin_prefetch(ap + k + 64, 0, 1);
        }
        #pragma unroll
        for (int i = 0; i < BM; ++i) {
            #pragma unroll
            for (int j = 0; j < BN; ++j) {
                acc[i][j] = __builtin_amdgcn_wmma_f32_16x16x4_f32(
                                false, a[i], false, b[j], (short)0, acc[i][j],
                                false, false);
            }
        }
    }

    const int rr = (lane >> 4) << 3;        // +0 or +8 row offset within tile
    #pragma unroll
    for (int j = 0; j < BN; ++j) {
        const int ccol = n0 + j * 16 + r;
        float bv = 0.f;
        if (EPI != EPI_NONE) bv = bias[ccol];
        #pragma unroll
        for (int i = 0; i < BM; ++i) {
            #pragma unroll
            for (int q = 0; q < 8; ++q) {
                float v = acc[i][j][q];
                if (EPI == EPI_BIAS) {
                    v += bv;
                } else if (EPI == EPI_BIAS_GELU) {
                    v += bv;
                    v = 0.5f * v * (1.f + erff(v * 0.70710678118654752f));
                } else if (EPI == EPI_BIAS_SOFTPLUS) {
                    v += bv;
                    v = (v > 20.f) ? v : log1pf(__expf(v));
                }
                C[(size_t)(m0 + i * 16 + rr + q) * ldc + ccol] = v;
            }
        }
    }
}

// ---------------------------------------------------------------------------
// Elementwise / row kernels
// ---------------------------------------------------------------------------
__global__ __launch_bounds__(256)
void rmsnorm_kernel(const float* __restrict__ x, const float* __restrict__ w,
                    float* __restrict__ y)
{
    __shared__ float sd[256];
    const int row = blockIdx.x, c = threadIdx.x;
    float v = x[(size_t)row * DIM + c];
    sd[c] = v * v;
    __syncthreads();
    for (int s = 128; s > 0; s >>= 1) {
        if (c < s) sd[c] += sd[c + s];
        __syncthreads();
    }
    float inv = rsqrtf(sd[0] * (1.f / DIM) + 1e-6f);
    y[(size_t)row * DIM + c] = v * inv * w[c];
}

// depthwise conv(k=3,pad=1) over L, then LayerNorm(256) + SiLU, fused per token
__global__ __launch_bounds__(256)
void localconv_ln_silu_kernel(const float* __restrict__ xn,
                              const float* __restrict__ cw, const float* __restrict__ cb,
                              const float* __restrict__ gw, const float* __restrict__ gb,
                              float* __restrict__ xc)
{
    __shared__ float s1[256], s2[256];
    const int t = blockIdx.x, c = threadIdx.x;
    const int b = t / SEQ, l = t % SEQ;
    float acc = cb[c];
    #pragma unroll
    for (int j = 0; j < 3; ++j) {
        int ll = l - 1 + j;
        if (ll >= 0 && ll < SEQ)
            acc += cw[c * 3 + j] * xn[((size_t)(b * SEQ + ll)) * DIM + c];
    }
    s1[c] = acc; s2[c] = acc * acc;
    __syncthreads();
    for (int s = 128; s > 0; s >>= 1) {
        if (c < s) { s1[c] += s1[c + s]; s2[c] += s2[c + s]; }
        __syncthreads();
    }
    float m   = s1[0] * (1.f / DIM);
    float var = s2[0] * (1.f / DIM) - m * m;
    float n   = (acc - m) * rsqrtf(var + 1e-5f) * gw[c] + gb[c];
    xc[(size_t)t * DIM + c] = silu_f(n);
}

__global__ __launch_bounds__(256)
void reverse_seq_kernel(const float* __restrict__ src, float* __restrict__ dst)
{
    const int t = blockIdx.x, c = threadIdx.x;
    const int b = t / SEQ, l = t % SEQ;
    dst[(size_t)t * DIM + c] = src[((size_t)b * SEQ + (SEQ - 1 - l)) * DIM + c];
}

// causal depthwise conv (k=4, left pad 3) + SiLU over xm = xz[..., :512]
__global__ __launch_bounds__(256)
void causal_conv_silu_kernel(const float* __restrict__ xz,
                             const float* __restrict__ w, const float* __restrict__ bias,
                             float* __restrict__ xmc)
{
    const size_t idx = (size_t)blockIdx.x * blockDim.x + threadIdx.x;  // NT*512
    const int    d   = (int)(idx % DINNER);
    const size_t row = idx / DINNER;
    const int    l   = (int)(row % SEQ);
    const size_t b0  = row - l;
    float acc = bias[d];
    #pragma unroll
    for (int j = 0; j < 4; ++j) {
        int ll = l - 3 + j;
        if (ll >= 0)
            acc += w[d * 4 + j] * xz[(b0 + ll) * (2 * DINNER) + d];
    }
    xmc[idx] = silu_f(acc);
}

// Selective scan: one wave per (b,d), one state per lane (D_STATE == wave32).
__global__ __launch_bounds__(256)
void ssm_scan_kernel(const float* __restrict__ delta, const float* __restrict__ xmc,
                     const float* __restrict__ xdbl, const float* __restrict__ A_log,
                     float* __restrict__ y)
{
    const int lane = threadIdx.x & 31;
    const int wid  = blockIdx.x * (blockDim.x >> 5) + (threadIdx.x >> 5); // b*512 + d
    const int b = wid >> 9;
    const int d = wid & (DINNER - 1);
    const float A = -__expf(A_log[d * DSTATE + lane]);

    const float* dp = delta + (size_t)b * SEQ * DINNER + d;
    const float* xp = xmc   + (size_t)b * SEQ * DINNER + d;
    const float* bp = xdbl  + (size_t)b * SEQ * 80 + DTRANK + lane;
    const float* cp = xdbl  + (size_t)b * SEQ * 80 + DTRANK + DSTATE + lane;
    float*       yp = y     + (size_t)b * SEQ * DINNER + d;

    float h = 0.f;
    for (int t = 0; t < SEQ; ++t) {
        const float dt = dp[(size_t)t * DINNER];
        const float xt = xp[(size_t)t * DINNER];
        const float Bt = bp[(size_t)t * 80];
        const float Ct = cp[(size_t)t * 80];
        h = h * __expf(dt * A) + dt * xt * Bt;
        float v = h * Ct;
        #pragma unroll
        for (int off = 16; off > 0; off >>= 1)
            v += __shfl_xor(v, off, 32);
        if (lane == 0) yp[(size_t)t * DINNER] = v;
    }
}

// y = (y + xm*D) * silu(z),  z = xz[..., 512:]
__global__ __launch_bounds__(256)
void gate_kernel(float* __restrict__ y, const float* __restrict__ xmc,
                 const float* __restrict__ xz, const float* __restrict__ Dv)
{
    const size_t idx = (size_t)blockIdx.x * blockDim.x + threadIdx.x;  // NT*512
    const int    d   = (int)(idx % DINNER);
    const size_t row = idx / DINNER;
    const float  z   = xz[row * (2 * DINNER) + DINNER + d];
    y[idx] = (y[idx] + xmc[idx] * Dv[d]) * silu_f(z);
}

// xs[:, :, :256] = mf + xn ; xs[:, :, 256:] = mb reversed-in-L + xn
__global__ __launch_bounds__(256)
void build_xs_kernel(const float* __restrict__ mf, const float* __restrict__ mb,
                     const float* __restrict__ xn, float* __restrict__ xs)
{
    const size_t idx = (size_t)blockIdx.x * blockDim.x + threadIdx.x;  // NT*512
    const int    c   = (int)(idx % (2 * DIM));
    const size_t row = idx / (2 * DIM);
    const int    l   = (int)(row % SEQ);
    const size_t b0  = row - l;
    float v;
    if (c < DIM) {
        v = mf[row * DIM + c] + xn[row * DIM + c];
    } else {
        const size_t rr = b0 + (SEQ - 1 - l);
        v = mb[rr * DIM + (c - DIM)] + xn[row * DIM + (c - DIM)];
    }
    xs[idx] = v;
}

// in-place LayerNorm over 512 channels
__global__ __launch_bounds__(256)
void ln512_kernel(float* __restrict__ xs, const float* __restrict__ w,
                  const float* __restrict__ b)
{
    __shared__ float s1[256], s2[256];
    const int row = blockIdx.x, tid = threadIdx.x;
    float v0 = xs[(size_t)row * 512 + tid];
    float v1 = xs[(size_t)row * 512 + 256 + tid];
    s1[tid] = v0 + v1; s2[tid] = v0 * v0 + v1 * v1;
    __syncthreads();
    for (int s = 128; s > 0; s >>= 1) {
        if (tid < s) { s1[tid] += s1[tid + s]; s2[tid] += s2[tid + s]; }
        __syncthreads();
    }
    float m   = s1[0] * (1.f / 512.f);
    float var = s2[0] * (1.f / 512.f) - m * m;
    float inv = rsqrtf(var + 1e-5f);
    xs[(size_t)row * 512 + tid]       = (v0 - m) * inv * w[tid] + b[tid];
    xs[(size_t)row * 512 + 256 + tid] = (v1 - m) * inv * w[256 + tid] + b[256 + tid];
}

// grouped conv (512->256, 2 in-ch per group, k=3, pad=1) + SiLU
__global__ __launch_bounds__(256)
void postconv_silu_kernel(const float* __restrict__ xsn, const float* __restrict__ w,
                          const float* __restrict__ bias, float* __restrict__ outp)
{
    const int t = blockIdx.x, o = threadIdx.x;
    const int b = t / SEQ, l = t % SEQ;
    float acc = bias[o];
    #pragma unroll
    for (int j = 0; j < 3; ++j) {
        int ll = l - 1 + j;
        if (ll < 0 || ll >= SEQ) continue;
        const float* rowp = xsn + ((size_t)(b * SEQ + ll)) * 512;
        acc += w[o * 6 + j]     * rowp[2 * o];
        acc += w[o * 6 + 3 + j] * rowp[2 * o + 1];
    }
    outp[(size_t)t * DIM + o] = silu_f(acc);
}

// out = x + softmax(logits)[0]*mixer + softmax(logits)[1]*mlp
__global__ __launch_bounds__(256)
void combine_kernel(const float* __restrict__ x, const float* __restrict__ mixer,
                    const float* __restrict__ xmlp, const float* __restrict__ logits,
                    float* __restrict__ out)
{
    const size_t idx = (size_t)blockIdx.x * blockDim.x + threadIdx.x;  // NT*256
    const float l0 = logits[0], l1 = logits[1];
    const float mx = fmaxf(l0, l1);
    const float e0 = __expf(l0 - mx), e1 = __expf(l1 - mx);
    const float inv = 1.f / (e0 + e1);
    out[idx] = x[idx] + (e0 * inv) * mixer[idx] + (e1 * inv) * xmlp[idx];
}

// ---------------------------------------------------------------------------
// Host orchestration
// ---------------------------------------------------------------------------
template<int EPI, int BM, int BN>
static void launch_gemm(const float* A, int lda, const float* W, int ldw,
                        const float* bias, float* C, int ldc,
                        int M, int N, int K, hipStream_t s)
{
    const int waves  = (M / (16 * BM)) * (N / (16 * BN));
    const int blocks = (waves + 7) / 8;     // 8 waves (256 threads) per block
    gemm_wmma_f32<EPI, BM, BN><<<blocks, 256, 0, s>>>(A, lda, W, ldw, bias,
                                                      C, ldc, M, N, K);
}

extern "C" void kernel_launch(void* const* d_in, const int* in_sizes, int n_in,
                              void* d_out, int out_size, void* d_ws, size_t ws_size,
                              hipStream_t stream)
{
    const float* x         = (const float*)d_in[0];
    const float* rms_w     = (const float*)d_in[1];
    const float* lconv_w   = (const float*)d_in[2];
    const float* lconv_b   = (const float*)d_in[3];
    const float* lnc_w     = (const float*)d_in[4];
    const float* lnc_b     = (const float*)d_in[5];
    const float* in_proj_w = (const float*)d_in[6];
    const float* conv1d_w  = (const float*)d_in[7];
    const float* conv1d_b  = (const float*)d_in[8];
    const float* xproj_w   = (const float*)d_in[9];
    const float* dtproj_w  = (const float*)d_in[10];
    const float* dtproj_b  = (const float*)d_in[11];
    const float* A_log     = (const float*)d_in[12];
    const float* ssm_D     = (const float*)d_in[13];
    const float* outproj_w = (const float*)d_in[14];
    const float* lnp_w     = (const float*)d_in[15];
    const float* lnp_b     = (const float*)d_in[16];
    const float* pconv_w   = (const float*)d_in[17];
    const float* pconv_b   = (const float*)d_in[18];
    const float* mlp_w1    = (const float*)d_in[19];
    const float* mlp_b1    = (const float*)d_in[20];
    const float* mlp_w2    = (const float*)d_in[21];
    const float* mlp_b2    = (const float*)d_in[22];
    const float* blogits   = (const float*)d_in[23];
    float* out = (float*)d_out;

    // workspace layout (floats)
    float* ws    = (float*)d_ws;
    float* xn    = ws;                                  // NT*256
    float* xc    = xn    + (size_t)NT * DIM;            // NT*256
    float* xcrev = xc    + (size_t)NT * DIM;            // NT*256
    float* xz    = xcrev + (size_t)NT * DIM;            // NT*1024 (reused as MLP hidden)
    float* xmc   = xz    + (size_t)NT * 2 * DINNER;     // NT*512
    float* xdbl  = xmc   + (size_t)NT * DINNER;         // NT*80
    float* dlt   = xdbl  + (size_t)NT * 80;             // NT*512
    float* ysc   = dlt   + (size_t)NT * DINNER;         // NT*512
    float* mf    = ysc   + (size_t)NT * DINNER;         // NT*256
    float* mb    = mf    + (size_t)NT * DIM;            // NT*256
    float* xs    = mb    + (size_t)NT * DIM;            // NT*512 (LN in place)
    float* mixer = xs    + (size_t)NT * 2 * DIM;        // NT*256
    float* xmlp  = mixer + (size_t)NT * DIM;            // NT*256

    // front-end
    rmsnorm_kernel<<<NT, 256, 0, stream>>>(x, rms_w, xn);
    localconv_ln_silu_kernel<<<NT, 256, 0, stream>>>(xn, lconv_w, lconv_b, lnc_w, lnc_b, xc);
    reverse_seq_kernel<<<NT, 256, 0, stream>>>(xc, xcrev);

    // bidirectional mamba (shared weights), sequential through shared scratch
    for (int dir = 0; dir < 2; ++dir) {
        const float* src  = dir ? xcrev : xc;
        float*       mout = dir ? mb : mf;

        launch_gemm<EPI_NONE, 4, 4>(src, DIM, in_proj_w, DIM, nullptr,
                                    xz, 2 * DINNER, NT, 2 * DINNER, DIM, stream);
        causal_conv_silu_kernel<<<(NT * DINNER) / 256, 256, 0, stream>>>(
            xz, conv1d_w, conv1d_b, xmc);
        launch_gemm<EPI_NONE, 4, 1>(xmc, DINNER, xproj_w, DINNER, nullptr,
                                    xdbl, 80, NT, 80, DINNER, stream);
        launch_gemm<EPI_BIAS_SOFTPLUS, 4, 4>(xdbl, 80, dtproj_w, DTRANK, dtproj_b,
                                             dlt, DINNER, NT, DINNER, DTRANK, stream);
        ssm_scan_kernel<<<(BATCH * DINNER) / 8, 256, 0, stream>>>(
            dlt, xmc, xdbl, A_log, ysc);
        gate_kernel<<<(NT * DINNER) / 256, 256, 0, stream>>>(ysc, xmc, xz, ssm_D);
        launch_gemm<EPI_NONE, 4, 4>(ysc, DINNER, outproj_w, DINNER, nullptr,
                                    mout, DIM, NT, DIM, DINNER, stream);
    }

    // back-end
    build_xs_kernel<<<(NT * 2 * DIM) / 256, 256, 0, stream>>>(mf, mb, xn, xs);
    ln512_kernel<<<NT, 256, 0, stream>>>(xs, lnp_w, lnp_b);
    postconv_silu_kernel<<<NT, 256, 0, stream>>>(xs, pconv_w, pconv_b, mixer);

    launch_gemm<EPI_BIAS_GELU, 4, 4>(xn, DIM, mlp_w1, DIM, mlp_b1,
                                     xz, 4 * DIM, NT, 4 * DIM, DIM, stream);  // reuse xz
    launch_gemm<EPI_BIAS, 4, 4>(xz, 4 * DIM, mlp_w2, 4 * DIM, mlp_b2,
                                xmlp, DIM, NT, DIM, 4 * DIM, stream);

    combine_kernel<<<NT, 256, 0, stream>>>(x, mixer, xmlp, blogits, out);
}